// BERTNet_51007031607489
// MI455X (gfx1250) — compile-verified
//
#include <hip/hip_runtime.h>
#include <hip/hip_bf16.h>
#include <math.h>
#include <stdint.h>

// MI455X / gfx1250 fused kernel for BERT pair-MLP head.
// Stage 1: segment-mean pool subwords -> word_emb (+ one zero row) in d_ws.
// Stage 2: per 16-pair block: async-gather 16x1536 A-tile straight into LDS
//          (global_load_async_to_lds_b128, ASYNCcnt), fp32 WMMA
//          (v_wmma_f32_16x16x4_f32) GEMM vs w1 (1536x300), tanh, tiny
//          second GEMM vs w2 (300x4) + softmax.

typedef float v2f __attribute__((ext_vector_type(2)));
typedef float v8f __attribute__((ext_vector_type(8)));

#define BB 32
#define SS 512
#define DD 768
#define WW 256
#define PP 1024
#define KK 1536          // 2*D
#define NH 300           // hidden units
#define NTILES 19        // ceil(300/16) -> columns padded to 304
#define ASTRIDE 1540     // 1536 + 4 pad: 16 rows land on 16 distinct banks
#define XSTRIDE 304
#define ZROW (BB * WW)   // index of the all-zeros embedding row

// ---------------------------------------------------------------------------
// Kernel 1: word_emb[b,w,:] = mean(h[b, start : start+len, :]), len in {1,2}
// grid = B*W + 1 blocks (last block writes the zero row), 192 thr, float4.
// ---------------------------------------------------------------------------
__global__ __launch_bounds__(192)
void bert_pool_kernel(const float* __restrict__ h,
                      const int* __restrict__ word_start,
                      const int* __restrict__ word_len,
                      float* __restrict__ word_emb) {
  const int bw = blockIdx.x;          // b*W + w, or ZROW
  const int d4 = threadIdx.x << 2;

  if (bw == ZROW) {                   // dedicated zero embedding row
    *(float4*)(word_emb + (size_t)bw * DD + d4) = make_float4(0.f, 0.f, 0.f, 0.f);
    return;
  }

  const int b  = bw >> 8;             // W = 256
  const int st = word_start[bw];
  const int ln = word_len[bw];

  const float* r0 = h + ((size_t)b * SS + st) * DD + d4;
  float4 v = *(const float4*)r0;
  if (ln == 2) {
    float4 u = *(const float4*)(r0 + DD);
    v.x = (v.x + u.x) * 0.5f;
    v.y = (v.y + u.y) * 0.5f;
    v.z = (v.z + u.z) * 0.5f;
    v.w = (v.w + u.w) * 0.5f;
  }
  *(float4*)(word_emb + (size_t)bw * DD + d4) = v;
}

// ---------------------------------------------------------------------------
// Kernel 2: fused async-gather + GEMM1(tanh) + GEMM2 + softmax.
// grid = (B*P)/16 = 2048 blocks, 256 threads = 8 wave32.
// Dynamic LDS: A-tile 16x1540 f32 + X-tile 16x304 f32 = 118016 bytes.
// ---------------------------------------------------------------------------
__global__ __launch_bounds__(256)
void bert_pair_mlp_kernel(const float* __restrict__ word_emb,
                          const int* __restrict__ pair_idx,
                          const float* __restrict__ w1,
                          const float* __restrict__ b1,
                          const float* __restrict__ w2,
                          const float* __restrict__ b2,
                          float* __restrict__ out) {
  extern __shared__ float smem[];
  float* At = smem;                       // [16][ASTRIDE]
  float* Xt = smem + 16 * ASTRIDE;        // [16][XSTRIDE]

  const int tid  = threadIdx.x;
  const int gp0  = blockIdx.x << 4;       // first global pair-row of block
  const int wave = tid >> 5;              // wave32
  const int lane = tid & 31;

  // ---- async gather of A-tile directly into LDS.
  // 192 chunks = 16 rows x 2 sides x 6 chunks of 512B (32 lanes x 16B).
  // Invalid pair index (-1) is redirected to the zero row (ZROW).
  for (int c = wave; c < 192; c += 8) {
    const int row  = c / 12;
    const int rem  = c - row * 12;
    const int side = rem / 6;
    const int ch   = rem - side * 6;      // 128-float chunk within one side
    const int gp   = gp0 + row;
    const int bi   = gp >> 10;            // P = 1024
    const int pi   = pair_idx[((size_t)gp << 1) + side];
    const int wrow = (pi >= 0) ? (bi * WW + pi) : ZROW;

    const float* gsrc = word_emb + (size_t)wrow * DD + (ch << 7);
    unsigned long long gaddr =
        (unsigned long long)(uintptr_t)gsrc + (unsigned)(lane << 4);
    unsigned laddr =
        (unsigned)(uintptr_t)(At + row * ASTRIDE + side * DD + (ch << 7)) +
        (unsigned)(lane << 4);
    asm volatile("global_load_async_to_lds_b128 %0, %1, off"
                 :: "v"(laddr), "v"(gaddr) : "memory");
  }
  asm volatile("s_wait_asynccnt 0x0" ::: "memory");
  __syncthreads();

  const int lhi = lane >> 4;              // 0/1 half-wave
  const int llo = lane & 15;

  // ---- layer 1: x = tanh(A @ w1 + b1), one 16x16 tile per wave iteration.
  for (int nt = wave; nt < NTILES; nt += 8) {
    const int nbase = nt << 4;
    const int n  = nbase + llo;           // output column this lane owns
    const int nc = (n < NH) ? n : 0;      // clamped (padded cols discarded)

    v8f acc = {0.f, 0.f, 0.f, 0.f, 0.f, 0.f, 0.f, 0.f};
    // ISA A/B layout: VGPR v, lane-high h -> K = k0 + 2h + v
    const float* ap = At + llo * ASTRIDE + 2 * lhi;
    const float* bp = w1 + (size_t)(2 * lhi) * NH + nc;

#pragma unroll 4
    for (int k0 = 0; k0 < KK; k0 += 4) {
      v2f a = *(const v2f*)(ap + k0);
      v2f b;
      b.x = bp[(size_t)(k0)     * NH];
      b.y = bp[(size_t)(k0 + 1) * NH];
      acc = __builtin_amdgcn_wmma_f32_16x16x4_f32(
          /*neg_a=*/false, a, /*neg_b=*/false, b,
          /*c_mod=*/(short)0, acc, /*reuse_a=*/false, /*reuse_b=*/false);
    }

    const float bias = (n < NH) ? b1[n] : 0.f;
    const bool  live = (n < NH);
    // C/D layout: VGPR j, lane-high h -> row M = j + 8h, col N = lane&15
#pragma unroll
    for (int j = 0; j < 8; ++j) {
      const int row = j + (lhi << 3);
      Xt[row * XSTRIDE + n] = live ? tanhf(acc[j] + bias) : 0.f;
    }
  }
  __syncthreads();

  // ---- layer 2 + softmax: one thread per output row (16 rows, 4 logits).
  if (tid < 16) {
    const float* xr = Xt + tid * XSTRIDE;
    float l0 = b2[0], l1 = b2[1], l2 = b2[2], l3 = b2[3];
    for (int k = 0; k < NH; ++k) {
      const float xv = xr[k];
      const float4 wv = *(const float4*)(w2 + (size_t)k * 4);
      l0 += xv * wv.x;
      l1 += xv * wv.y;
      l2 += xv * wv.z;
      l3 += xv * wv.w;
    }
    const float m  = fmaxf(fmaxf(l0, l1), fmaxf(l2, l3));
    const float e0 = expf(l0 - m), e1 = expf(l1 - m);
    const float e2 = expf(l2 - m), e3 = expf(l3 - m);
    const float inv = 1.f / (e0 + e1 + e2 + e3);
    float* op = out + ((size_t)(gp0 + tid) << 2);
    op[0] = e0 * inv;
    op[1] = e1 * inv;
    op[2] = e2 * inv;
    op[3] = e3 * inv;
  }
}

// ---------------------------------------------------------------------------
// Host-side launcher. Input order (setup_inputs dict):
//   0:h  1:w1  2:b1  3:w2  4:b2  5:word_start  6:word_len  7:pair_idx
// d_out: float[B*P*4].  d_ws: word_emb scratch ((B*W+1)*D floats ~ 25 MB).
// ---------------------------------------------------------------------------
extern "C" void kernel_launch(void* const* d_in, const int* in_sizes, int n_in,
                              void* d_out, int out_size, void* d_ws, size_t ws_size,
                              hipStream_t stream) {
  const float* h          = (const float*)d_in[0];
  const float* w1         = (const float*)d_in[1];
  const float* b1         = (const float*)d_in[2];
  const float* w2         = (const float*)d_in[3];
  const float* b2         = (const float*)d_in[4];
  const int*   word_start = (const int*)d_in[5];
  const int*   word_len   = (const int*)d_in[6];
  const int*   pair_idx   = (const int*)d_in[7];

  float* word_emb = (float*)d_ws;                 // (B*W + 1) * D floats
  float* out      = (float*)d_out;

  bert_pool_kernel<<<BB * WW + 1, 192, 0, stream>>>(h, word_start, word_len,
                                                    word_emb);

  const size_t smem_bytes = (size_t)(16 * ASTRIDE + 16 * XSTRIDE) * sizeof(float);
  bert_pair_mlp_kernel<<<(BB * PP) / 16, 256, smem_bytes, stream>>>(
      word_emb, pair_idx, w1, b1, w2, b2, out);
}